// SymmetricContraction_24361054503258
// MI455X (gfx1250) — compile-verified
//
#include <hip/hip_runtime.h>
#include <stdint.h>

typedef __attribute__((ext_vector_type(2))) float v2f;
typedef __attribute__((ext_vector_type(8))) float v8f;

// ---------------- problem constants ----------------
// irreps: 0e(1) 1o(3) 2e(5) 3o(7), D=16
// muls: order3 {5,8,10,10}, order2 {2,3,4,4}, order1 {1,1,1,1}
constexpr int NP3 = 160;   // sum m*di = 149 padded to 160 (10 N-blocks)
constexpr int NP2 = 64;    // sum = 59 padded to 64 (4 N-blocks)
constexpr int NP1 = 16;    // sum = 16 (1 N-block)
constexpr int K3  = 4096;  // D^3
constexpr int K2  = 256;   // D^2
constexpr int K1  = 16;    // D

constexpr int SZ_U3 = K3 * NP3;          // 655360
constexpr int SZ_U2 = K2 * NP2;          // 16384
constexpr int SZ_U1 = K1 * NP1;          // 256
constexpr int OFF_U3 = 0;
constexpr int OFF_U2 = OFF_U3 + SZ_U3;
constexpr int OFF_U1 = OFF_U2 + SZ_U2;
constexpr int SZ_W3 = 64 * 33 * 128;     // per-species order-3 weights
constexpr int SZ_W2 = 64 * 13 * 128;
constexpr int SZ_W1 = 64 * 4 * 128;
constexpr int OFF_W3 = OFF_U1 + SZ_U1;
constexpr int OFF_W2 = OFF_W3 + SZ_W3;
constexpr int OFF_W1 = OFF_W2 + SZ_W2;

// ---------------- deterministic stand-in RNG for U basis ----------------
__device__ inline uint32_t hash3(uint32_t a, uint32_t b, uint32_t c) {
    uint32_t x = a * 0x9E3779B1u ^ b * 0x85EBCA77u ^ c * 0xC2B2AE3Du;
    x ^= x >> 16; x *= 0x7FEB352Du;
    x ^= x >> 15; x *= 0x846CA68Bu;
    x ^= x >> 16;
    return x;
}

__device__ inline void emit_u(float* base, int order, int q, int p, int NP) {
    // valid (non-pad) column range per order
    int lim = (order == 3) ? 149 : (order == 2) ? 59 : 16;
    float val = 0.0f;
    if (p < lim) {
        uint32_t hm = hash3(0xA000u + order, (uint32_t)q, (uint32_t)(p * 3 + 2));
        if (hm < 429496730u) {  // ~10% density mask
            uint32_t h1 = hash3(0xA000u + order, (uint32_t)q, (uint32_t)(p * 3));
            uint32_t h2 = hash3(0xA000u + order, (uint32_t)q, (uint32_t)(p * 3 + 1));
            float u1 = ((float)h1 + 0.5f) * 2.3283064e-10f;
            float u2 = (float)h2 * 2.3283064e-10f;
            val = sqrtf(-2.0f * logf(u1)) * cosf(6.2831853f * u2) / (float)order;
        }
    }
    // K-pair swizzle: element (q,p) at [(q>>1)*NP + p]*2 + (q&1)  -> b64 loads
    base[(size_t)((q >> 1) * NP + p) * 2 + (q & 1)] = val;
}

__global__ void fill_u_kernel(float* __restrict__ ws) {
    int t = blockIdx.x * blockDim.x + threadIdx.x;
    if (t < SZ_U3) {
        emit_u(ws + OFF_U3, 3, t / NP3, t % NP3, NP3);
    } else if (t < SZ_U3 + SZ_U2) {
        int r = t - SZ_U3;
        emit_u(ws + OFF_U2, 2, r / NP2, r % NP2, NP2);
    } else if (t < SZ_U3 + SZ_U2 + SZ_U1) {
        int r = t - SZ_U3 - SZ_U2;
        emit_u(ws + OFF_U1, 1, r / NP1, r % NP1, NP1);
    }
}

// ---------------- per-species weight gather: w[s,k,c] = embed[s,:] @ W[:,k,c] ----------------
__global__ void species_weights(const float* __restrict__ emb, const float* __restrict__ W,
                                float* __restrict__ out, int mul, int kbase, int ktot) {
    int t = blockIdx.x * blockDim.x + threadIdx.x;
    int total = 64 * mul * 128;
    if (t >= total) return;
    int c = t & 127;
    int k = (t >> 7) % mul;
    int s = t / (mul * 128);
    float sum = 0.0f;
    #pragma unroll 4
    for (int e = 0; e < 32; ++e)
        sum += emb[s * 32 + e] * W[(e * mul + k) * 128 + c];
    out[((size_t)s * ktot + kbase + k) * 128 + c] = sum;
}

// ---------------- WMMA wrapper ----------------
__device__ inline v8f wmma4(v2f a, v2f b, v8f c) {
    // D = A(16x4 f32) * B(4x16 f32) + C(16x16 f32)
    return __builtin_amdgcn_wmma_f32_16x16x4_f32(false, a, false, b, (short)0, c,
                                                 false, false);
}

// ---------------- main fused kernel ----------------
__global__ __launch_bounds__(128) void symcon_main(const float* __restrict__ x,
                                                   const int* __restrict__ index,
                                                   const float* __restrict__ ws,
                                                   float* __restrict__ out) {
    __shared__ float xt[16 * 17];       // x tile [n][d], padded stride
    __shared__ float Pl[32 * 16 * 2];   // monomial panel chunk: [qpair][n][2]
    __shared__ float Ol[240 * 16];      // staged Out[p][n]

    const float* U3 = ws + OFF_U3;
    const float* U2 = ws + OFF_U2;
    const float* U1 = ws + OFF_U1;
    const float* W3 = ws + OFF_W3;
    const float* W2 = ws + OFF_W2;
    const float* W1 = ws + OFF_W1;

    const int T = threadIdx.x;
    const int tile = blockIdx.x;          // 2048 tiles of 16 columns
    const int b = tile >> 3;
    const int c0 = (tile & 7) << 4;
    const int s = index[b];

    // load x tile: n = col in tile, contiguous 256 floats
    for (int t = T; t < 256; t += 128)
        xt[(t >> 4) * 17 + (t & 15)] = x[(size_t)tile * 256 + t];
    __syncthreads();

    const int wave = T >> 5;
    const int lane = T & 31;
    const int h = lane >> 4;      // lane half
    const int nl = lane & 15;     // N / M index within fragment

    v8f acc[4];
    #pragma unroll
    for (int sl = 0; sl < 4; ++sl) {
        v8f z = {0.f, 0.f, 0.f, 0.f, 0.f, 0.f, 0.f, 0.f};
        acc[sl] = z;
    }

    const int bn = T & 15;          // build: column
    const int qb = (T >> 4) * 8;    // build: 8 q's per thread

    // ---- heavy order-3 GEMM: K = 4096, A built cooperatively per 64-chunk ----
    for (int kc = 0; kc < K3; kc += 64) {
        #pragma unroll
        for (int u = 0; u < 8; ++u) {
            int ql = qb + u;
            int q = kc + ql;
            int a1 = q >> 8, a2 = (q >> 4) & 15, j = q & 15;
            float v = xt[bn * 17 + a1] * xt[bn * 17 + a2] * xt[bn * 17 + j];
            Pl[(ql >> 1) * 32 + bn * 2 + (ql & 1)] = v;
        }
        __syncthreads();

        v2f af[16];
        #pragma unroll
        for (int kk = 0; kk < 16; ++kk)
            af[kk] = *(const v2f*)&Pl[((2 * kk + h) * 16 + nl) * 2];

        #pragma unroll
        for (int sl = 0; sl < 4; ++sl) {
            int bb = wave + sl * 4;
            if (bb < 10) {  // wave-uniform branch; EXEC stays all-ones for WMMA
                const float* Bb = U3 + ((size_t)(kc >> 1) * NP3 + (bb * 16 + nl)) * 2;
                #pragma unroll
                for (int kk = 0; kk < 16; ++kk) {
                    v2f bf = *(const v2f*)&Bb[(size_t)(2 * kk + h) * (NP3 * 2)];
                    acc[sl] = wmma4(af[kk], bf, acc[sl]);
                }
            }
        }
        __syncthreads();
    }

    // ---- light tails: order-2 (K=256) and order-1 (K=16), A built on the fly ----
    #pragma unroll
    for (int sl = 0; sl < 4; ++sl) {
        int bb = wave + sl * 4;
        if (bb >= 10 && bb < 14) {
            int p0 = (bb - 10) * 16;
            for (int kc = 0; kc < K2; kc += 4) {
                int q0 = kc + 2 * h;
                v2f af2;
                af2.x = xt[nl * 17 + (q0 >> 4)] * xt[nl * 17 + (q0 & 15)];
                int q1 = q0 + 1;
                af2.y = xt[nl * 17 + (q1 >> 4)] * xt[nl * 17 + (q1 & 15)];
                v2f bf = *(const v2f*)&U2[((size_t)((kc >> 1) + h) * NP2 + p0 + nl) * 2];
                acc[sl] = wmma4(af2, bf, acc[sl]);
            }
        } else if (bb == 14) {
            for (int kc = 0; kc < K1; kc += 4) {
                int q0 = kc + 2 * h;
                v2f af2;
                af2.x = xt[nl * 17 + q0];
                af2.y = xt[nl * 17 + q0 + 1];
                v2f bf = *(const v2f*)&U1[((size_t)((kc >> 1) + h) * NP1 + nl) * 2];
                acc[sl] = wmma4(af2, bf, acc[sl]);
            }
        }
    }

    // ---- stage Out[p][n] in LDS ----
    #pragma unroll
    for (int sl = 0; sl < 4; ++sl) {
        int bb = wave + sl * 4;
        if (bb < 15) {
            #pragma unroll
            for (int r = 0; r < 8; ++r)
                Ol[(bb * 16 + nl) * 16 + (r + 8 * h)] = acc[sl][r];
        }
    }
    __syncthreads();

    // ---- fused epilogue: out[n,i] = sum_k w[s,k,c]*Out[p(ir,k,i)][n] ----
    const int muls3[4] = {5, 8, 10, 10};
    const int muls2[4] = {2, 3, 4, 4};
    const int dims[4]  = {1, 3, 5, 7};
    const int kb3[4]   = {0, 5, 13, 23};
    const int kb2[4]   = {0, 2, 5, 9};
    const int pb3[4]   = {0, 5, 29, 79};
    const int pb2[4]   = {160, 162, 171, 191};
    const int pb1[4]   = {224, 225, 228, 233};
    const int ib[4]    = {0, 1, 4, 9};

    for (int oi = T; oi < 256; oi += 128) {
        int n = oi >> 4, i = oi & 15;
        int ir = (i >= 9) ? 3 : (i >= 4) ? 2 : (i >= 1) ? 1 : 0;
        int il = i - ib[ir];
        int di = dims[ir];
        float sum = 0.0f;
        const float* w3row = W3 + ((size_t)s * 33 + kb3[ir]) * 128 + c0 + n;
        for (int k = 0; k < muls3[ir]; ++k)
            sum += w3row[(size_t)k * 128] * Ol[(pb3[ir] + k * di + il) * 16 + n];
        const float* w2row = W2 + ((size_t)s * 13 + kb2[ir]) * 128 + c0 + n;
        for (int k = 0; k < muls2[ir]; ++k)
            sum += w2row[(size_t)k * 128] * Ol[(pb2[ir] + k * di + il) * 16 + n];
        sum += W1[((size_t)s * 4 + ir) * 128 + c0 + n] * Ol[(pb1[ir] + il) * 16 + n];
        out[((size_t)tile * 16 + n) * 16 + i] = sum;
    }
}

// ---------------- host launcher ----------------
extern "C" void kernel_launch(void* const* d_in, const int* in_sizes, int n_in,
                              void* d_out, int out_size, void* d_ws, size_t ws_size,
                              hipStream_t stream) {
    const float* x     = (const float*)d_in[0];
    const int*   index = (const int*)d_in[1];
    const float* embed = (const float*)d_in[2];
    float* ws = (float*)d_ws;

    // 1) materialize scaled+swizzled U matrices (deterministic every call)
    int totU = SZ_U3 + SZ_U2 + SZ_U1;
    fill_u_kernel<<<(totU + 255) / 256, 256, 0, stream>>>(ws);

    // 2) per-species gathered path weights, 12 (order, irrep) segments
    const int muls3[4] = {5, 8, 10, 10};
    const int muls2[4] = {2, 3, 4, 4};
    int arg = 3, kb = 0;
    for (int ir = 0; ir < 4; ++ir) {
        int m = muls3[ir], tot = 64 * m * 128;
        species_weights<<<(tot + 255) / 256, 256, 0, stream>>>(
            embed, (const float*)d_in[arg++], ws + OFF_W3, m, kb, 33);
        kb += m;
    }
    kb = 0;
    for (int ir = 0; ir < 4; ++ir) {
        int m = muls2[ir], tot = 64 * m * 128;
        species_weights<<<(tot + 255) / 256, 256, 0, stream>>>(
            embed, (const float*)d_in[arg++], ws + OFF_W2, m, kb, 13);
        kb += m;
    }
    kb = 0;
    for (int ir = 0; ir < 4; ++ir) {
        int m = 1, tot = 64 * m * 128;
        species_weights<<<(tot + 255) / 256, 256, 0, stream>>>(
            embed, (const float*)d_in[arg++], ws + OFF_W1, m, kb, 4);
        kb += m;
    }

    // 3) fused polynomial-GEMM + epilogue
    symcon_main<<<2048, 128, 0, stream>>>(x, index, ws, (float*)d_out);
}